// SSD_249108103514
// MI455X (gfx1250) — compile-verified
//
#include <hip/hip_runtime.h>
#include <hip/hip_bf16.h>
#include <stdint.h>

#define BNUM 4
#define PNUM 32768
#define CNUM 81
#define TOPK 200
#define CONF_T 0.01f
#define NMS_T 0.45f
#define THREADS 256
#define NWAVES (THREADS / 32)
#define CHUNK (PNUM / THREADS) /* 128 contiguous elements per thread */

typedef unsigned int u32x4 __attribute__((ext_vector_type(4)));
typedef int i32x8 __attribute__((ext_vector_type(8)));
typedef int i32x4 __attribute__((ext_vector_type(4)));

// float -> order-preserving uint (descending float == descending uint)
__device__ __forceinline__ unsigned f2s(float f) {
    unsigned u = __float_as_uint(f);
    return (u & 0x80000000u) ? ~u : (u | 0x80000000u);
}
__device__ __forceinline__ float s2f(unsigned s) {
    unsigned u = (s & 0x80000000u) ? (s & 0x7FFFFFFFu) : ~s;
    return __uint_as_float(u);
}

// ---------------------------------------------------------------------------
// Kernel A: zero the entire output (class 0 rows and non-kept rows stay 0).
// ---------------------------------------------------------------------------
__global__ void ssd_zero_out(float* out, int n) {
    int i = blockIdx.x * blockDim.x + threadIdx.x;
    if (i < n) out[i] = 0.0f;
}

// ---------------------------------------------------------------------------
// Kernel B: per-anchor box decode + softmax statistics (row max, 1/sum).
// One thread per (b,p); streams ~45 MB once -> HBM bound (~2 us @ 23.3 TB/s).
// ---------------------------------------------------------------------------
__global__ void ssd_decode_stats(const float* __restrict__ locs,
                                 const float* __restrict__ logits,
                                 const float* __restrict__ priors,
                                 float* __restrict__ boxes,
                                 float* __restrict__ rmax,
                                 float* __restrict__ rinv) {
    int t = blockIdx.x * blockDim.x + threadIdx.x;
    if (t >= BNUM * PNUM) return;
    int p = t & (PNUM - 1);

    float4 l  = ((const float4*)locs)[t];
    float4 pr = ((const float4*)priors)[p];
    float cx = pr.x + l.x * 0.1f * pr.z;
    float cy = pr.y + l.y * 0.1f * pr.w;
    float w  = pr.z * __expf(l.z * 0.2f);
    float h  = pr.w * __expf(l.w * 0.2f);
    float x1 = cx - 0.5f * w;
    float y1 = cy - 0.5f * h;
    ((float4*)boxes)[t] = make_float4(x1, y1, x1 + w, y1 + h);

    const float* row = logits + (size_t)t * CNUM;
    float m = -3.402823466e38f;
    for (int c = 0; c < CNUM; ++c) m = fmaxf(m, row[c]);
    float s = 0.0f;
    for (int c = 0; c < CNUM; ++c) s += __expf(row[c] - m);
    rmax[t] = m;
    rinv[t] = 1.0f / s;
}

// ---------------------------------------------------------------------------
// Kernel C: one workgroup per (batch, foreground class).
//   1. TDM DMA of the strided logits column (32768 x 4B, stride 324B) to LDS
//   2. softmax prob + conf mask -> sortable uint, in place in LDS (128 KB)
//   3. 4-pass radix select with per-wave histogram replicas (contention /8)
//   4. index-ordered compaction (exact jax top_k tie-break), bitonic sort 256
//   5. barrier-free single-wave NMS (registers + shfl/ballot), scatter out
// ---------------------------------------------------------------------------
__global__ void __launch_bounds__(THREADS)
ssd_topk_nms(const float* __restrict__ logits,
             const float* __restrict__ boxes,
             const float* __restrict__ rmax,
             const float* __restrict__ rinv,
             float* __restrict__ out) {
    const int tid  = threadIdx.x;
    const int ccol = blockIdx.x + 1;  // logit column == output class (skip bg 0)
    const int b    = blockIdx.y;

    __shared__ unsigned s_vals[PNUM];              // 128 KB: logits -> sortable keys
    __shared__ unsigned s_hist[NWAVES * 256];      // per-wave histogram replicas
    __shared__ unsigned s_hsum[256];
    __shared__ unsigned s_cntA[THREADS];
    __shared__ unsigned s_cntB[THREADS];
    __shared__ unsigned long long s_key[THREADS];
    __shared__ float s_score[THREADS];
    __shared__ float s_bx[TOPK * 4];
    __shared__ int s_valid[THREADS];
    __shared__ int s_kept[THREADS];
    __shared__ int s_pos[TOPK];
    __shared__ int s_b0, s_b1, s_bG;

#if __has_builtin(__builtin_amdgcn_tensor_load_to_lds)
    // ---- Tensor Data Mover: DMA column ccol of logits[b] into LDS ----------
    if (tid < 32) {  // wave 0 only; TDM is wave-level, EXEC-ignored
        unsigned long long ga =
            (unsigned long long)(uintptr_t)(logits + (size_t)b * PNUM * CNUM + ccol);
        unsigned lds = (unsigned)(uintptr_t)(&s_vals[0]);
        u32x4 g0;
        g0[0] = 1u;                                   // count=1 (valid D#)
        g0[1] = lds;                                  // lds_addr
        g0[2] = (unsigned)(ga & 0xFFFFFFFFu);         // global_addr[31:0]
        g0[3] = (unsigned)((ga >> 32) & 0x1FFFFFFu) | (2u << 30); // addr hi | type=2
        i32x8 g1;
        g1[0] = (int)(2u << 16);        // data_size=4B; no mask/pad/barrier
        g1[1] = (int)(81u << 16);       // tensor_dim0 = 81 (row length)
        g1[2] = (int)0x80000000u;       // tensor_dim1 = 32768
        g1[3] = (int)0x00010000u;       // tile_dim0 = 1
        g1[4] = (int)0x00008000u;       // tile_dim1 = 32768
        g1[5] = 81;                     // tensor_dim0_stride = 81 elements
        g1[6] = 0;
        g1[7] = 0;
        i32x4 g2 = {0, 0, 0, 0};        // D# groups 2/3 unused (2D tensor)
        i32x4 g3 = {0, 0, 0, 0};
        i32x8 g4 = {0, 0, 0, 0, 0, 0, 0, 0}; // extended group (6-arg toolchain)
        __builtin_amdgcn_tensor_load_to_lds(g0, g1, g2, g3, g4, 0);
        __builtin_amdgcn_s_wait_tensorcnt(0);
    }
    __syncthreads();
#endif

    // ---- masked softmax prob -> sortable key, in place --------------------
    for (int i = tid; i < PNUM; i += THREADS) {
#if __has_builtin(__builtin_amdgcn_tensor_load_to_lds)
        float l = __uint_as_float(s_vals[i]);
#else
        float l = logits[((size_t)b * PNUM + i) * CNUM + ccol];
#endif
        float pr = __expf(l - rmax[b * PNUM + i]) * rinv[b * PNUM + i];
        float mv = (pr > CONF_T) ? pr : -1.0f;   // reference's masked scores
        s_vals[i] = f2s(mv);
    }
    __syncthreads();

    // ---- 4-pass radix select: threshold T with rank semantics -------------
    const int wave = tid >> 5;
    unsigned pref = 0, pmask = 0;
    int want = TOPK;
    for (int shift = 24; shift >= 0; shift -= 8) {
        for (int r = tid; r < NWAVES * 256; r += THREADS) s_hist[r] = 0;
        __syncthreads();
        for (int i = tid; i < PNUM; i += THREADS) {
            unsigned v = s_vals[i];
            if ((v & pmask) == pref)
                atomicAdd(&s_hist[(wave << 8) + ((v >> shift) & 255u)], 1u);
        }
        __syncthreads();
        {   // reduce replicas: thread d sums its digit across the 8 waves
            unsigned s = 0;
            for (int w = 0; w < NWAVES; ++w) s += s_hist[(w << 8) + tid];
            s_hsum[tid] = s;
        }
        __syncthreads();
        if (tid == 0) {
            unsigned cum = 0; int sel = 0;
            for (int dig = 255; dig >= 0; --dig) {
                unsigned c = s_hsum[dig];
                if (cum + c >= (unsigned)want) { sel = dig; break; }
                cum += c;
            }
            s_b0 = sel;
            s_b1 = want - (int)cum;
        }
        __syncthreads();
        pref |= ((unsigned)s_b0) << shift;
        pmask |= 0xFFu << shift;
        want = s_b1;            // # of ==T elements still needed (lowest idx)
        __syncthreads();
    }
    const unsigned T = pref;

    // ---- index-ordered compaction (exact jax top_k tie-break) -------------
    const int cb = tid * CHUNK;
    unsigned cg = 0, ce = 0;
    for (int i = 0; i < CHUNK; ++i) {
        unsigned v = s_vals[cb + i];
        cg += (v > T); ce += (v == T);
    }
    s_cntA[tid] = cg; s_cntB[tid] = ce;
    __syncthreads();
    if (tid == 0) {
        unsigned run = 0;
        for (int t2 = 0; t2 < THREADS; ++t2) { unsigned x = s_cntA[t2]; s_cntA[t2] = run; run += x; }
        s_bG = (int)run;    // G = #(v > T); G + want == TOPK
        run = 0;
        for (int t2 = 0; t2 < THREADS; ++t2) { unsigned x = s_cntB[t2]; s_cntB[t2] = run; run += x; }
    }
    __syncthreads();
    {
        int gpos = (int)s_cntA[tid], epos = (int)s_cntB[tid], G = s_bG;
        for (int i = 0; i < CHUNK; ++i) {
            unsigned v = s_vals[cb + i];
            int slot = -1;
            if (v > T) slot = gpos++;
            else if (v == T) { int r = epos++; if (r < want) slot = G + r; }
            if (slot >= 0)
                s_key[slot] = (((unsigned long long)v) << 16) |
                              (unsigned long long)(32767 - (cb + i)); // idx asc tiebreak
        }
    }
    if (tid >= TOPK) s_key[tid] = 0ull;  // pad sentinels (sort to the back)
    __syncthreads();

    // ---- bitonic sort, descending, 256 keys / 256 threads -----------------
    for (int k = 2; k <= THREADS; k <<= 1) {
        for (int j = k >> 1; j > 0; j >>= 1) {
            int ixj = tid ^ j;
            if (ixj > tid) {
                unsigned long long a = s_key[tid], c2 = s_key[ixj];
                bool descSeg = (tid & k) == 0;
                bool sw = descSeg ? (a < c2) : (a > c2);
                if (sw) { s_key[tid] = c2; s_key[ixj] = a; }
            }
            __syncthreads();
        }
    }

    // ---- unpack top-200: score, validity, gathered decoded box ------------
    if (tid < TOPK) {
        unsigned long long key = s_key[tid];
        unsigned sval = (unsigned)(key >> 16);
        int idx = 32767 - (int)(key & 0xFFFFull);
        float val = s2f(sval);
        s_score[tid] = val;
        s_valid[tid] = (val > CONF_T) ? 1 : 0;
        const float* bp = boxes + ((size_t)b * PNUM + idx) * 4;
        s_bx[tid * 4 + 0] = bp[0];
        s_bx[tid * 4 + 1] = bp[1];
        s_bx[tid * 4 + 2] = bp[2];
        s_bx[tid * 4 + 3] = bp[3];
    }
    s_kept[tid] = 0;
    __syncthreads();

    // ---- barrier-free greedy NMS in wave 0 (registers + shfl + ballot) ----
    // Slot i owned by lane (i&31), local index j=i>>5 (7 slots/lane max).
    if (tid < 32) {
        const int lane = tid;
        float cx1[7], cy1[7], cx2[7], cy2[7];
        unsigned lvalid = 0, lkept = 0;
#define SSD_LOADSLOT(J)                                             \
        { int i = (J) * 32 + lane; cx1[J]=0; cy1[J]=0; cx2[J]=0; cy2[J]=0; \
          if (i < TOPK) {                                           \
              cx1[J] = s_bx[i*4+0]; cy1[J] = s_bx[i*4+1];           \
              cx2[J] = s_bx[i*4+2]; cy2[J] = s_bx[i*4+3];           \
              if (s_valid[i]) lvalid |= 1u << (J); } }
        SSD_LOADSLOT(0) SSD_LOADSLOT(1) SSD_LOADSLOT(2) SSD_LOADSLOT(3)
        SSD_LOADSLOT(4) SSD_LOADSLOT(5) SSD_LOADSLOT(6)
#undef SSD_LOADSLOT

        for (int i = 0; i < TOPK; ++i) {
            const int owner = i & 31;
            const int j = i >> 5;           // uniform across the wave
            float bx1v, by1v, bx2v, by2v;
            switch (j) {                     // keep slot regs out of scratch
                case 0: bx1v=cx1[0]; by1v=cy1[0]; bx2v=cx2[0]; by2v=cy2[0]; break;
                case 1: bx1v=cx1[1]; by1v=cy1[1]; bx2v=cx2[1]; by2v=cy2[1]; break;
                case 2: bx1v=cx1[2]; by1v=cy1[2]; bx2v=cx2[2]; by2v=cy2[2]; break;
                case 3: bx1v=cx1[3]; by1v=cy1[3]; bx2v=cx2[3]; by2v=cy2[3]; break;
                case 4: bx1v=cx1[4]; by1v=cy1[4]; bx2v=cx2[4]; by2v=cy2[4]; break;
                case 5: bx1v=cx1[5]; by1v=cy1[5]; bx2v=cx2[5]; by2v=cy2[5]; break;
                default:bx1v=cx1[6]; by1v=cy1[6]; bx2v=cx2[6]; by2v=cy2[6]; break;
            }
            bx1v = __shfl(bx1v, owner, 32);
            by1v = __shfl(by1v, owner, 32);
            bx2v = __shfl(bx2v, owner, 32);
            by2v = __shfl(by2v, owner, 32);
            float ai = (bx2v - bx1v) * (by2v - by1v);

            int f = 0;
            unsigned m = lkept;              // test only my kept boxes
            while (m) {
                int j2 = __builtin_ctz(m); m &= m - 1;
                float dx1, dy1, dx2, dy2;
                switch (j2) {
                    case 0: dx1=cx1[0]; dy1=cy1[0]; dx2=cx2[0]; dy2=cy2[0]; break;
                    case 1: dx1=cx1[1]; dy1=cy1[1]; dx2=cx2[1]; dy2=cy2[1]; break;
                    case 2: dx1=cx1[2]; dy1=cy1[2]; dx2=cx2[2]; dy2=cy2[2]; break;
                    case 3: dx1=cx1[3]; dy1=cy1[3]; dx2=cx2[3]; dy2=cy2[3]; break;
                    case 4: dx1=cx1[4]; dy1=cy1[4]; dx2=cx2[4]; dy2=cy2[4]; break;
                    case 5: dx1=cx1[5]; dy1=cy1[5]; dx2=cx2[5]; dy2=cy2[5]; break;
                    default:dx1=cx1[6]; dy1=cy1[6]; dx2=cx2[6]; dy2=cy2[6]; break;
                }
                float xx1 = fmaxf(bx1v, dx1), yy1 = fmaxf(by1v, dy1);
                float xx2 = fminf(bx2v, dx2), yy2 = fminf(by2v, dy2);
                float iw = fmaxf(xx2 - xx1, 0.0f), ih = fmaxf(yy2 - yy1, 0.0f);
                float inter = iw * ih;
                float a2 = (dx2 - dx1) * (dy2 - dy1);
                if (inter / (ai + a2 - inter) > NMS_T) { f = 1; break; }
            }
            unsigned long long anyb = __ballot(f);
            if (lane == owner && ((lvalid >> j) & 1u) && anyb == 0ull)
                lkept |= 1u << j;
        }
        for (int j = 0; j < 7; ++j) {
            int i = j * 32 + lane;
            if (i < TOPK) s_kept[i] = (lkept >> j) & 1;
        }
    }
    __syncthreads();

    // ---- compact kept rows to the front, scatter to out -------------------
    if (tid == 0) {
        int cnt = 0;
        for (int i = 0; i < TOPK; ++i) s_pos[i] = s_kept[i] ? cnt++ : -1;
    }
    __syncthreads();
    if (tid < TOPK && s_kept[tid]) {
        size_t base = (((size_t)b * CNUM + ccol) * TOPK + s_pos[tid]) * 5;
        out[base + 0] = s_score[tid];
        out[base + 1] = s_bx[tid * 4 + 0];
        out[base + 2] = s_bx[tid * 4 + 1];
        out[base + 3] = s_bx[tid * 4 + 2];
        out[base + 4] = s_bx[tid * 4 + 3];
    }
}

// ---------------------------------------------------------------------------
extern "C" void kernel_launch(void* const* d_in, const int* in_sizes, int n_in,
                              void* d_out, int out_size, void* d_ws, size_t ws_size,
                              hipStream_t stream) {
    const float* locs   = (const float*)d_in[0];
    const float* logits = (const float*)d_in[1];
    const float* priors = (const float*)d_in[2];
    float* out = (float*)d_out;

    // workspace: boxes[B*P*4] | rmax[B*P] | rinv[B*P]  (3 MB total)
    float* ws    = (float*)d_ws;
    float* boxes = ws;
    float* rmax  = ws + (size_t)BNUM * PNUM * 4;
    float* rinv  = rmax + (size_t)BNUM * PNUM;

    ssd_zero_out<<<(out_size + THREADS - 1) / THREADS, THREADS, 0, stream>>>(out, out_size);
    ssd_decode_stats<<<(BNUM * PNUM + THREADS - 1) / THREADS, THREADS, 0, stream>>>(
        locs, logits, priors, boxes, rmax, rinv);
    dim3 grid(CNUM - 1, BNUM);  // 80 fg classes x 4 images = 320 workgroups
    ssd_topk_nms<<<grid, THREADS, 0, stream>>>(logits, boxes, rmax, rinv, out);
}